// CausalStdMeanScaler_15582141350258
// MI455X (gfx1250) — compile-verified
//
#include <hip/hip_runtime.h>
#include <stdint.h>

// Problem constants (from reference): B=64, V=256, T=4096
#define T_DIM 4096
#define ROWS  (64 * 256)          // 16384 independent rows
#define WAVES_PER_BLOCK 8
#define LANES 32
#define CHUNK 128                 // elements per wave per iteration (4 per lane)
#define NCHUNK (T_DIM / CHUNK)    // 32 serial chunks per row
#define NBUF 3                    // prefetch distance 2 -> triple buffer

typedef __attribute__((ext_vector_type(4))) float v4f;  // native clang vector

// ---- CDNA5 async global->LDS copy (tracked by ASYNCcnt) --------------------
// LDS destination address comes from the vdst VGPR; generic pointer low 32 bits == LDS offset.
__device__ __forceinline__ uint32_t lds_off32(const void* p) {
  return (uint32_t)(uintptr_t)p;
}

__device__ __forceinline__ void async_copy16(const float* __restrict__ gsrc,
                                             float* __restrict__ ldst) {
  // Streaming input: read-once, mark non-temporal so it doesn't occupy L2.
  asm volatile("global_load_async_to_lds_b128 %0, %1, off th:TH_LOAD_NT"
               :
               : "v"(lds_off32(ldst)), "v"((uint64_t)(uintptr_t)gsrc)
               : "memory");
}

// ---- wave32 inclusive scan via ds_bpermute ---------------------------------
__device__ __forceinline__ float bperm_up(float v, int lane, int d) {
  int r = __builtin_amdgcn_ds_bpermute((lane - d) << 2, __float_as_int(v));
  return __int_as_float(r);   // garbage when lane<d, masked below
}

__device__ __forceinline__ float wave_incl_scan(float v, int lane) {
#pragma unroll
  for (int d = 1; d < LANES; d <<= 1) {
    float u = bperm_up(v, lane, d);
    if (lane >= d) v += u;
  }
  return v;
}

__device__ __forceinline__ float bcast_lane31(float v) {
  return __int_as_float(__builtin_amdgcn_readlane(__float_as_int(v), 31));
}

// ---------------------------------------------------------------------------
__global__ __launch_bounds__(WAVES_PER_BLOCK * LANES)
void CausalStdMeanScaler_kernel(const float* __restrict__ data,
                                const float* __restrict__ mask,
                                const float* __restrict__ wts,
                                float* __restrict__ out_scaled,
                                float* __restrict__ out_mean,
                                float* __restrict__ out_scale) {
  // triple-buffered LDS stage: [wave][buf][stream: data/mask/weights][128]
  __shared__ __align__(16) float lds[WAVES_PER_BLOCK][NBUF][3][CHUNK];

  const int lane = (int)(threadIdx.x & (LANES - 1));
  const int wave = (int)(threadIdx.x >> 5);
  const int row  = (int)blockIdx.x * WAVES_PER_BLOCK + wave;
  if (row >= ROWS) return;

  const size_t rbase = (size_t)row * T_DIM;
  const int loff = lane * 4;

  // prime the pipeline: chunks 0,1 -> buffers 0,1 (3 async b128 each)
#pragma unroll
  for (int p = 0; p < 2; ++p) {
    const size_t g = rbase + (size_t)p * CHUNK + loff;
    async_copy16(data + g, &lds[wave][p][0][loff]);
    async_copy16(mask + g, &lds[wave][p][1][loff]);
    async_copy16(wts  + g, &lds[wave][p][2][loff]);
  }

  float carryW = 0.0f, carryS = 0.0f, carryM = 0.0f;

  int buf = 0;
  for (int c = 0; c < NCHUNK; ++c) {
    if (c + 2 < NCHUNK) {
      // issue chunk c+2 while chunks c (landing) and c+1 are in flight
      int nbuf = buf + 2; if (nbuf >= NBUF) nbuf -= NBUF;
      const size_t g = rbase + (size_t)(c + 2) * CHUNK + loff;
      async_copy16(data + g, &lds[wave][nbuf][0][loff]);
      async_copy16(mask + g, &lds[wave][nbuf][1][loff]);
      async_copy16(wts  + g, &lds[wave][nbuf][2][loff]);
      // up to 6 newer loads (c+1, c+2) may stay outstanding; chunk c is done
      asm volatile("s_wait_asynccnt 0x6" ::: "memory");
    } else if (c + 1 < NCHUNK) {
      asm volatile("s_wait_asynccnt 0x3" ::: "memory");
    } else {
      asm volatile("s_wait_asynccnt 0x0" ::: "memory");
    }

    const v4f xv = *(const v4f*)&lds[wave][buf][0][loff];
    const v4f pv = *(const v4f*)&lds[wave][buf][1][loff];
    const v4f wv = *(const v4f*)&lds[wave][buf][2][loff];

    float x[4]   = {xv.x, xv.y, xv.z, xv.w};
    float pm[4]  = {pv.x, pv.y, pv.z, pv.w};
    float wgt[4] = {wv.x, wv.y, wv.z, wv.w};

    // w = weights*mask ; y = w*x ; lane-local inclusive scans over 4 elems
    float w_[4], y_[4], lw[4], ly[4];
#pragma unroll
    for (int i = 0; i < 4; ++i) { w_[i] = wgt[i] * pm[i]; y_[i] = w_[i] * x[i]; }
    lw[0] = w_[0]; ly[0] = y_[0];
#pragma unroll
    for (int i = 1; i < 4; ++i) { lw[i] = lw[i-1] + w_[i]; ly[i] = ly[i-1] + y_[i]; }

    // wave scans of lane totals -> exclusive lane offsets (+ row carry)
    const float inclW = wave_incl_scan(lw[3], lane);
    const float inclS = wave_incl_scan(ly[3], lane);
    const float exclW = inclW - lw[3] + carryW;
    const float exclS = inclS - ly[3] + carryS;

    // means, shifted means, Welford increments
    float mean[4], Wt[4], inc[4];
    float pW = exclW, pS = exclS;      // prefix through previous element
#pragma unroll
    for (int i = 0; i < 4; ++i) {
      const float Wi = exclW + lw[i];
      const float Si = exclS + ly[i];
      const float m  = Si / fmaxf(Wi, 1.0f);
      const float mp = pS / fmaxf(pW, 1.0f);   // t=0 -> 0/1 = 0 (shifted pad)
      inc[i]  = (x[i] - mp) * (x[i] - m) * w_[i];
      mean[i] = m;
      Wt[i]   = Wi;
      pW = Wi; pS = Si;
    }

    // third scan: M2 = cumsum(increment)
    float li[4];
    li[0] = inc[0];
#pragma unroll
    for (int i = 1; i < 4; ++i) li[i] = li[i-1] + inc[i];
    const float inclM = wave_incl_scan(li[3], lane);
    const float exclM = inclM - li[3] + carryM;

    float z[4], sc[4];
#pragma unroll
    for (int i = 0; i < 4; ++i) {
      const float m2  = exclM + li[i];
      const float var = m2 / fmaxf(Wt[i] - 1.0f, 1.0f);   // Bessel
      sc[i] = sqrtf(var + 0.1f);                          // MINIMUM_SCALE
      z[i]  = (x[i] - mean[i]) / sc[i];
    }

    // streaming outputs: write-once, non-temporal stores (NT temporal hint)
    const size_t g = rbase + (size_t)c * CHUNK + loff;
    const v4f zv = {z[0], z[1], z[2], z[3]};
    const v4f mv = {mean[0], mean[1], mean[2], mean[3]};
    const v4f sv = {sc[0], sc[1], sc[2], sc[3]};
    __builtin_nontemporal_store(zv, (v4f*)(out_scaled + g));
    __builtin_nontemporal_store(mv, (v4f*)(out_mean + g));
    __builtin_nontemporal_store(sv, (v4f*)(out_scale + g));

    // propagate row carries from lane 31's inclusive totals
    carryW += bcast_lane31(inclW);
    carryS += bcast_lane31(inclS);
    carryM += bcast_lane31(inclM);

    if (++buf >= NBUF) buf = 0;
  }
}

extern "C" void kernel_launch(void* const* d_in, const int* in_sizes, int n_in,
                              void* d_out, int out_size, void* d_ws, size_t ws_size,
                              hipStream_t stream) {
  (void)in_sizes; (void)n_in; (void)d_ws; (void)ws_size; (void)out_size;

  const float* data = (const float*)d_in[0];   // data
  const float* mask = (const float*)d_in[1];   // padding_mask
  const float* wts  = (const float*)d_in[2];   // weights

  const size_t N = (size_t)ROWS * T_DIM;
  float* out_scaled = (float*)d_out;           // tuple order: scaled, means, scale
  float* out_mean   = out_scaled + N;
  float* out_scale  = out_mean + N;

  dim3 block(WAVES_PER_BLOCK * LANES);
  dim3 grid(ROWS / WAVES_PER_BLOCK);
  CausalStdMeanScaler_kernel<<<grid, block, 0, stream>>>(
      data, mask, wts, out_scaled, out_mean, out_scale);
}